// RandomGNNPositionalEncodings_8143257994110
// MI455X (gfx1250) — compile-verified
//
#include <hip/hip_runtime.h>

#define NN 20000
#define MS 30
#define EE 640000
#define HD 128
#define DD 512
#define EPSV 1e-5f

typedef __attribute__((ext_vector_type(2))) float v2f;
typedef __attribute__((ext_vector_type(8))) float v8f;
typedef int v4i __attribute__((vector_size(16)));
typedef __attribute__((address_space(1))) v4i* gbl_v4i_p;
typedef __attribute__((address_space(3))) v4i* lds_v4i_p;

#if defined(__gfx1250__) && __has_builtin(__builtin_amdgcn_global_load_async_to_lds_b128) && \
    __has_builtin(__builtin_amdgcn_s_wait_asynccnt)
#define HAVE_ASYNC_LDS 1
#else
#define HAVE_ASYNC_LDS 0
#endif

// ---------------- utility kernels ----------------
__global__ void k_zero_f32(float* p, int n) {
    int i = blockIdx.x * blockDim.x + threadIdx.x;
    if (i < n) p[i] = 0.f;
}
__global__ void k_zero_u32(unsigned* p, int n) {
    int i = blockIdx.x * blockDim.x + threadIdx.x;
    if (i < n) p[i] = 0u;
}
// zero one 128-wide slab of the (N,512) concat buffer (slab = base ptr already offset)
__global__ void k_zero_slab(float* slab) {
    int i = blockIdx.x * blockDim.x + threadIdx.x;  // over N*32 float4s
    if (i < NN * 32) {
        int n = i >> 5, c4 = (i & 31) * 4;
        *(float4*)&slab[(size_t)n * DD + c4] = make_float4(0.f, 0.f, 0.f, 0.f);
    }
}

// ---------------- graph preprocessing ----------------
__global__ void k_degree(const int* ei, unsigned* deg) {
    int e = blockIdx.x * blockDim.x + threadIdx.x;
    if (e < EE) atomicAdd(&deg[ei[EE + e]], 1u);
}
__global__ void k_dinv(const unsigned* deg, float* dinv) {
    int n = blockIdx.x * blockDim.x + threadIdx.x;
    if (n < NN) {
        unsigned d = deg[n];
        dinv[n] = d > 0u ? rsqrtf((float)d) : 0.f;
    }
}
__global__ void k_wedge(const int* ei, const float* dinv, float* we) {
    int e = blockIdx.x * blockDim.x + threadIdx.x;
    if (e < EE) we[e] = dinv[ei[e]] * dinv[ei[EE + e]];
}

// ---------------- conv0 path (C=1) ----------------
__global__ void k_init_x0(const float* Q, int s, float* X0) {
    int n = blockIdx.x * blockDim.x + threadIdx.x;
    if (n < NN) {
        X0[n * 4 + 0] = Q[n * MS + s];
        X0[n * 4 + 1] = 0.f;
        X0[n * 4 + 2] = 0.f;
        X0[n * 4 + 3] = 0.f;
    }
}
__global__ void k_prop_scalar(const int* ei, const float* we, float* X0, int hop) {
    int e = blockIdx.x * blockDim.x + threadIdx.x;
    if (e < EE) {
        int s = ei[e], d = ei[EE + e];
        unsafeAtomicAdd(&X0[d * 4 + hop + 1], we[e] * X0[s * 4 + hop]);
    }
}
// (N,4)@(4,128)+b -> LN(ln0) -> relu -> Xcat slab0.  One 128-thread block per node.
__global__ __launch_bounds__(128) void k_conv0(const float* X0, const float* W0, const float* b0,
                                               const float* g, const float* be, float* Xcat) {
    __shared__ float xs[4];
    __shared__ float rs[128], rq[128];
    int n = blockIdx.x, h = threadIdx.x;
    if (h < 4) xs[h] = X0[n * 4 + h];
    __syncthreads();
    float y = b0[h];
#pragma unroll
    for (int j = 0; j < 4; ++j) y += xs[j] * W0[j * HD + h];
    rs[h] = y;
    rq[h] = y * y;
    __syncthreads();
    for (int st = 64; st > 0; st >>= 1) {
        if (h < st) { rs[h] += rs[h + st]; rq[h] += rq[h + st]; }
        __syncthreads();
    }
    float mu = rs[0] * (1.f / HD);
    float var = rq[0] * (1.f / HD) - mu * mu;
    float o = fmaxf((y - mu) * rsqrtf(var + EPSV) * g[h] + be[h], 0.f);
    Xcat[(size_t)n * DD + h] = o;
}

// ---------------- 128-channel propagation: one wave per edge ----------------
__global__ __launch_bounds__(256) void k_prop_vec(const int* ei, const float* we,
                                                  const float* xin, float* xout) {
    int wave = threadIdx.x >> 5, lane = threadIdx.x & 31;
    int e = blockIdx.x * 8 + wave;
    if (e >= EE) return;
    int s = ei[e], d = ei[EE + e];
    float w = we[e];
    float4 v = *(const float4*)&xin[(size_t)s * DD + lane * 4];
    float* dp = &xout[(size_t)d * DD + lane * 4];
    unsafeAtomicAdd(dp + 0, w * v.x);
    unsafeAtomicAdd(dp + 1, w * v.y);
    unsafeAtomicAdd(dp + 2, w * v.z);
    unsafeAtomicAdd(dp + 3, w * v.w);
}

// ---------------- WMMA GEMM: Y = Xcat(N,512) @ W(512,128) + b, fused epilogue ----------------
// mode 0: relu(LN(Y))+x   mode 1: relu(Y)+x   mode 2: Y       -> written to Xcat slab0
#define LDA 516  // 512 + 4 pad: rows land on distinct LDS banks
#define LDY 132
__global__ __launch_bounds__(256) void k_tag_gemm(float* Xcat, const float* W, const float* bias,
                                                  const float* lng, const float* lnb, int mode) {
    __shared__ float As[16 * LDA];
    __shared__ float Ys[16 * LDY];
    __shared__ float rs[256], rq[256];
    __shared__ float rmean[16], rrstd[16];

    int tid = threadIdx.x;
    int n0 = blockIdx.x * 16;
    // stage 16x512 A tile into LDS
#if HAVE_ASYNC_LDS
    for (int i = tid; i < 16 * 128; i += 256) {
        int r = i >> 7, c4 = (i & 127) * 4;
        __builtin_amdgcn_global_load_async_to_lds_b128(
            (gbl_v4i_p)&Xcat[(size_t)(n0 + r) * DD + c4],
            (lds_v4i_p)&As[r * LDA + c4], 0, 0);
    }
    __builtin_amdgcn_s_wait_asynccnt(0);
#else
    for (int i = tid; i < 16 * 128; i += 256) {
        int r = i >> 7, c4 = (i & 127) * 4;
        float4 v = *(const float4*)&Xcat[(size_t)(n0 + r) * DD + c4];
        float* dp = &As[r * LDA + c4];
        dp[0] = v.x; dp[1] = v.y; dp[2] = v.z; dp[3] = v.w;
    }
#endif
    __syncthreads();

    int wave = tid >> 5, lane = tid & 31;
    int half = lane >> 4, l16 = lane & 15;
    int ncol = wave * 16 + l16;  // output column 0..127

    v8f acc = {};
    for (int k0 = 0; k0 < 512; k0 += 4) {
        int kk = k0 + 2 * half;
        v2f a, b;
        a.x = As[l16 * LDA + kk];
        a.y = As[l16 * LDA + kk + 1];
        b.x = W[kk * HD + ncol];
        b.y = W[(kk + 1) * HD + ncol];
        acc = __builtin_amdgcn_wmma_f32_16x16x4_f32(false, a, false, b, (short)0, acc, false, false);
    }
    float bv = bias[ncol];
#pragma unroll
    for (int r = 0; r < 8; ++r) {
        int m = r + 8 * half;
        Ys[m * LDY + ncol] = acc[r] + bv;
    }
    __syncthreads();

    if (mode == 0) {  // layernorm stats across 128 cols per row
        int row = tid & 15, seg = tid >> 4;  // 16 segments of 8 cols
        float s = 0.f, q = 0.f;
#pragma unroll
        for (int j = 0; j < 8; ++j) {
            float v = Ys[row * LDY + seg * 8 + j];
            s += v; q += v * v;
        }
        rs[seg * 16 + row] = s;
        rq[seg * 16 + row] = q;
        __syncthreads();
        if (tid < 16) {
            float ss = 0.f, qq = 0.f;
            for (int sg = 0; sg < 16; ++sg) { ss += rs[sg * 16 + tid]; qq += rq[sg * 16 + tid]; }
            float mu = ss * (1.f / HD);
            float var = qq * (1.f / HD) - mu * mu;
            rmean[tid] = mu;
            rrstd[tid] = rsqrtf(var + EPSV);
        }
        __syncthreads();
    }

    int m = tid & 15, cb = (tid >> 4) * 8;
#pragma unroll
    for (int j = 0; j < 8; ++j) {
        int c = cb + j;
        float v = Ys[m * LDY + c];
        float res = As[m * LDA + c];  // slab0 = residual input
        float o;
        if (mode == 0)      o = fmaxf((v - rmean[m]) * rrstd[m] * lng[c] + lnb[c], 0.f) + res;
        else if (mode == 1) o = fmaxf(v, 0.f) + res;
        else                o = v;
        Xcat[(size_t)(n0 + m) * DD + c] = o;
    }
}

// ---------------- WMMA projection: out += Xcat_slab0(N,128) @ Wp(128,512) + pb ----------------
#define LDP 132
__global__ __launch_bounds__(256) void k_proj(const float* Xcat, const float* Wp, const float* pb,
                                              float* outp) {
    __shared__ float As[16 * LDP];
    int tid = threadIdx.x;
    int n0 = blockIdx.x * 16;
    int gcol = blockIdx.y * 128;
#if HAVE_ASYNC_LDS
    for (int i = tid; i < 16 * 32; i += 256) {
        int r = i >> 5, c4 = (i & 31) * 4;
        __builtin_amdgcn_global_load_async_to_lds_b128(
            (gbl_v4i_p)&Xcat[(size_t)(n0 + r) * DD + c4],
            (lds_v4i_p)&As[r * LDP + c4], 0, 0);
    }
    __builtin_amdgcn_s_wait_asynccnt(0);
#else
    for (int i = tid; i < 16 * 32; i += 256) {
        int r = i >> 5, c4 = (i & 31) * 4;
        float4 v = *(const float4*)&Xcat[(size_t)(n0 + r) * DD + c4];
        float* dp = &As[r * LDP + c4];
        dp[0] = v.x; dp[1] = v.y; dp[2] = v.z; dp[3] = v.w;
    }
#endif
    __syncthreads();

    int wave = tid >> 5, lane = tid & 31;
    int half = lane >> 4, l16 = lane & 15;
    int ncol = gcol + wave * 16 + l16;  // 0..511

    v8f acc = {};
    for (int k0 = 0; k0 < 128; k0 += 4) {
        int kk = k0 + 2 * half;
        v2f a, b;
        a.x = As[l16 * LDP + kk];
        a.y = As[l16 * LDP + kk + 1];
        b.x = Wp[kk * DD + ncol];
        b.y = Wp[(kk + 1) * DD + ncol];
        acc = __builtin_amdgcn_wmma_f32_16x16x4_f32(false, a, false, b, (short)0, acc, false, false);
    }
    float bv = pb[ncol];
#pragma unroll
    for (int r = 0; r < 8; ++r) {
        int m = r + 8 * half;
        float* p = &outp[(size_t)(n0 + m) * DD + ncol];
        *p += acc[r] + bv;
    }
}

// ---------------- final batch-norm over nodes ----------------
__global__ __launch_bounds__(256) void k_bnstats(const float* outp, float* mu, float* var) {
    __shared__ float rs[256], rq[256];
    int d = blockIdx.x, tid = threadIdx.x;
    float s = 0.f, q = 0.f;
    for (int n = tid; n < NN; n += 256) {
        float v = outp[(size_t)n * DD + d] * (1.f / MS);
        s += v; q += v * v;
    }
    rs[tid] = s; rq[tid] = q;
    __syncthreads();
    for (int st = 128; st > 0; st >>= 1) {
        if (tid < st) { rs[tid] += rs[tid + st]; rq[tid] += rq[tid + st]; }
        __syncthreads();
    }
    if (tid == 0) {
        float m = rs[0] * (1.f / NN);
        mu[d] = m;
        var[d] = rq[0] * (1.f / NN) - m * m;
    }
}
__global__ void k_bnapply(float* outp, const float* mu, const float* var,
                          const float* g, const float* b) {
    int i = blockIdx.x * blockDim.x + threadIdx.x;
    if (i < NN * DD) {
        int d = i & (DD - 1);
        float v = outp[i] * (1.f / MS);
        outp[i] = (v - mu[d]) * rsqrtf(var[d] + EPSV) * g[d] + b[d];
    }
}

// ---------------- host orchestration ----------------
extern "C" void kernel_launch(void* const* d_in, const int* in_sizes, int n_in,
                              void* d_out, int out_size, void* d_ws, size_t ws_size,
                              hipStream_t stream) {
    const float* Q   = (const float*)d_in[0];
    const int*   ei  = (const int*)d_in[1];
    const float* W0  = (const float*)d_in[2];  const float* b0 = (const float*)d_in[3];
    const float* W1  = (const float*)d_in[4];  const float* b1 = (const float*)d_in[5];
    const float* W2  = (const float*)d_in[6];  const float* b2 = (const float*)d_in[7];
    const float* W3  = (const float*)d_in[8];  const float* b3 = (const float*)d_in[9];
    const float* l0g = (const float*)d_in[10]; const float* l0b = (const float*)d_in[11];
    const float* l1g = (const float*)d_in[12]; const float* l1b = (const float*)d_in[13];
    const float* Wp  = (const float*)d_in[14]; const float* pb = (const float*)d_in[15];
    const float* bng = (const float*)d_in[16]; const float* bnb = (const float*)d_in[17];
    float* outp = (float*)d_out;

    char* ws = (char*)d_ws;
    unsigned* deg = (unsigned*)ws;  ws += (size_t)NN * 4;
    float* dinv   = (float*)ws;     ws += (size_t)NN * 4;
    float* we     = (float*)ws;     ws += (size_t)EE * 4;
    float* X0     = (float*)ws;     ws += (size_t)NN * 4 * 4;
    float* Xcat   = (float*)ws;     ws += (size_t)NN * DD * 4;
    float* muv    = (float*)ws;     ws += (size_t)DD * 4;
    float* varv   = (float*)ws;

    const int TB = 256;
    k_zero_u32<<<(NN + TB - 1) / TB, TB, 0, stream>>>(deg, NN);
    k_zero_f32<<<(NN * DD + TB - 1) / TB, TB, 0, stream>>>(outp, NN * DD);
    k_degree<<<(EE + TB - 1) / TB, TB, 0, stream>>>(ei, deg);
    k_dinv<<<(NN + TB - 1) / TB, TB, 0, stream>>>(deg, dinv);
    k_wedge<<<(EE + TB - 1) / TB, TB, 0, stream>>>(ei, dinv, we);

    for (int s = 0; s < MS; ++s) {
        // conv0: scalar hops + tiny GEMM + LN + relu
        k_init_x0<<<(NN + TB - 1) / TB, TB, 0, stream>>>(Q, s, X0);
        for (int hop = 0; hop < 3; ++hop)
            k_prop_scalar<<<(EE + TB - 1) / TB, TB, 0, stream>>>(ei, we, X0, hop);
        k_conv0<<<NN, 128, 0, stream>>>(X0, W0, b0, l0g, l0b, Xcat);

        // conv1..conv3: 3 vector hops each + WMMA GEMM with fused epilogue
        const float* Ws[3]  = {W1, W2, W3};
        const float* bs[3]  = {b1, b2, b3};
        for (int cv = 0; cv < 3; ++cv) {
            for (int hop = 0; hop < 3; ++hop) {
                k_zero_slab<<<(NN * 32 + TB - 1) / TB, TB, 0, stream>>>(Xcat + (hop + 1) * HD);
                k_prop_vec<<<EE / 8, TB, 0, stream>>>(ei, we, Xcat + hop * HD, Xcat + (hop + 1) * HD);
            }
            k_tag_gemm<<<NN / 16, TB, 0, stream>>>(Xcat, Ws[cv], bs[cv], l1g, l1b, cv);
        }
        // projection, accumulated into d_out
        k_proj<<<dim3(NN / 16, 4), TB, 0, stream>>>(Xcat, Wp, pb, outp);
    }

    k_bnstats<<<DD, TB, 0, stream>>>(outp, muv, varv);
    k_bnapply<<<(NN * DD + TB - 1) / TB, TB, 0, stream>>>(outp, muv, varv, bng, bnb);
}